// VINNetwork_34428457844847
// MI455X (gfx1250) — compile-verified
//
#include <hip/hip_runtime.h>

// CDNA5 / gfx1250. wave32. WMMA F32_16x16x4_F32 for the channel matmul,
// K=10 value-iteration sweeps fully resident in LDS (one HBM pass).

typedef float v2f __attribute__((ext_vector_type(2)));
typedef float v8f __attribute__((ext_vector_type(8)));

#define IMG   512
#define TILE  32
#define KITER 10
#define HALO  KITER
#define REG   (TILE + 2 * HALO)     // 52
#define REG2  (REG * REG)           // 2704
#define NGRP  (REG2 / 16)           // 169 (exact)
#define INTR  (REG - 2)             // 50 interior span
#define NPOS  (INTR * INTR)         // 2500 updated positions
#define PPT   ((NPOS + 255) / 256)  // 10 positions per thread

__launch_bounds__(256)
__global__ void vin_kernel(const float* __restrict__ obs,
                           const float* __restrict__ Wp,
                           const float* __restrict__ bp,
                           float* __restrict__ out)
{
    __shared__ float sP[REG2];   // phi[...,0]  (p)
    __shared__ float sR[REG2];   // phi[...,1]  (rin)
    __shared__ float sO[REG2];   // phi[...,2]  (rout)
    __shared__ float sV0[REG2];  // v ping
    __shared__ float sV1[REG2];  // v pong

    const int t    = threadIdx.x;
    const int lane = t & 31;
    const int wave = t >> 5;
    const int n    = lane & 15;      // WMMA output column N (phi channel)
    const bool hi  = lane >= 16;

    const int w0 = blockIdx.x * TILE;
    const int h0 = blockIdx.y * TILE;
    const long imgBase = (long)blockIdx.z * (IMG * IMG);

    // ---- B matrix (4x16 f32): rows = {W[0,:], W[1,:], W[2,:], b} ----
    // lanes 0-15: VGPR0=K0, VGPR1=K1 ; lanes 16-31: VGPR0=K2, VGPR1=K3. N=lane&15.
    v2f bm; bm.x = 0.0f; bm.y = 0.0f;
    if (n < 3) {
        if (hi) { bm.x = Wp[6 + n]; bm.y = bp[n];     }
        else    { bm.x = Wp[n];     bm.y = Wp[3 + n]; }
    }

    // Zero v buffers (border ring must be exactly 0.0 == v0; also clears poison).
    for (int i = t; i < REG2; i += 256) { sV0[i] = 0.0f; sV1[i] = 0.0f; }

    // ---- Phase 1: phi = relu(obs*W + b) over 52x52 region via WMMA ----
    // A 16x4 f32: lanes 0-15 supply (c0,c1) of pixel M=lane; lanes 16-31 supply
    // (c2, 1.0) of pixel M=lane-16.  Manually software-pipelined: loads for
    // group g+8 are issued before the WMMA + LDS stores of group g.
    // Loads are UNconditional from clamped coords (no exec-masked branches).
    int rA = (wave * 16 + n) / REG;      // coords of this lane's A pixel
    int cA = (wave * 16 + n) % REG;
    int rB = (wave * 16) / REG;          // coords of group base (store side)
    int cB = (wave * 16) % REG;
    const int chOff = hi ? 2 : 0;

    // Prologue: load A for g = wave.
    v2f am;
    {
        const int h = h0 + rA - HALO, w = w0 + cA - HALO;
        const bool in = ((unsigned)h < (unsigned)IMG) & ((unsigned)w < (unsigned)IMG);
        const int hc = min(max(h, 0), IMG - 1), wc = min(max(w, 0), IMG - 1);
        const long off = (imgBase + (long)hc * IMG + wc) * 3;
        const float x0 = obs[off + chOff];
        const float x1 = obs[off + 1];
        am.x = in ? x0 : 0.0f;
        am.y = hi ? 1.0f : (in ? x1 : 0.0f);
    }

    for (int g = wave; g < NGRP; g += 8) {
        // --- issue next group's loads (group stride 16*8 = 128 = 2*REG + 24) ---
        rA += 2; cA += 24; if (cA >= REG) { cA -= REG; ++rA; }
        const int hn = h0 + rA - HALO, wn = w0 + cA - HALO;
        const bool inn = ((unsigned)hn < (unsigned)IMG) & ((unsigned)wn < (unsigned)IMG);
        const int hcn = min(max(hn, 0), IMG - 1), wcn = min(max(wn, 0), IMG - 1);
        const long offn = (imgBase + (long)hcn * IMG + wcn) * 3;
        const float nx0 = obs[offn + chOff];
        const float nx1 = obs[offn + 1];

        // --- WMMA for current group ---
        v8f acc = {0.f, 0.f, 0.f, 0.f, 0.f, 0.f, 0.f, 0.f};
        acc = __builtin_amdgcn_wmma_f32_16x16x4_f32(
                  false, am, false, bm, (short)0, acc, false, false);

        // --- store current group's D to LDS planes (no divisions) ---
        // D layout: VGPR r8 -> row M = r8 + (hi?8:0), column N = lane&15.
        if (n < 3) {
            float* plane = (n == 0) ? sP : (n == 1) ? sR : sO;
            int rr = rB, cc = cB;
            if (hi) { cc += 8; if (cc >= REG) { cc -= REG; ++rr; } }
            int idx2 = rr * REG + cc;
            #pragma unroll
            for (int r8 = 0; r8 < 8; ++r8) {
                const int hh = h0 + rr - HALO, ww = w0 + cc - HALO;
                const bool in2 = ((unsigned)hh < (unsigned)IMG) &
                                 ((unsigned)ww < (unsigned)IMG);
                plane[idx2] = in2 ? fmaxf(acc[r8], 0.0f) : 0.0f;
                ++cc; ++idx2;
                if (cc == REG) { cc = 0; ++rr; idx2 = rr * REG; }
            }
        }
        rB += 2; cB += 24; if (cB >= REG) { cB -= REG; ++rB; }

        // --- commit next A ---
        am.x = inn ? nx0 : 0.0f;
        am.y = hi ? 1.0f : (inn ? nx1 : 0.0f);
    }
    __syncthreads();

    // ---- Phase 2: hoist per-position constants into registers; sweep tt=0 ----
    float pc[PPT], ro[PPT], rUp[PPT], rDn[PPT], rLt[PPT], rRt[PPT], vs[PPT];
    int   ic[PPT];
    unsigned vmask = 0;

    #pragma unroll
    for (int k = 0; k < PPT; ++k) {
        const int idx = t + k * 256;
        const bool ok = idx < NPOS;
        const int ii  = ok ? idx : 0;
        const int r = 1 + ii / INTR, c = 1 + ii % INTR;
        const int i0 = r * REG + c;
        ic[k]  = i0;
        pc[k]  = sP[i0];
        ro[k]  = sO[i0];
        rUp[k] = sR[i0 + REG];   // neighbor (h+1, w)
        rDn[k] = sR[i0 - REG];   // neighbor (h-1, w)
        rLt[k] = sR[i0 + 1];     // neighbor (h, w+1)
        rRt[k] = sR[i0 - 1];     // neighbor (h, w-1)
        const int h = h0 + r - HALO, w = w0 + c - HALO;
        const bool valid = ok & ((unsigned)h < (unsigned)IMG) &
                                ((unsigned)w < (unsigned)IMG);
        if (valid) vmask |= (1u << k);
        // Sweep tt=0: v == 0 everywhere -> m = max(r_dirs) - rout; v = m.
        const float m0 = fmaxf(fmaxf(rUp[k], rDn[k]), fmaxf(rLt[k], rRt[k])) - ro[k];
        const float v  = valid ? m0 : 0.0f;
        vs[k] = v;
        if (ok) sV0[i0] = v;
    }
    __syncthreads();

    // ---- Sweeps tt=1..9, ping-pong in LDS (4 LDS loads + 1 store per pos) ----
    #pragma unroll
    for (int it = 1; it < KITER; ++it) {
        float* __restrict__ vin  = (it & 1) ? sV0 : sV1;
        float* __restrict__ vout = (it & 1) ? sV1 : sV0;
        #pragma unroll
        for (int k = 0; k < PPT; ++k) {
            const int i0 = ic[k];
            const float vU = vin[i0 + REG];
            const float vD = vin[i0 - REG];
            const float vL = vin[i0 + 1];
            const float vR = vin[i0 - 1];
            float m = fmaxf(fmaxf(fmaf(pc[k], vU, rUp[k]),
                                  fmaf(pc[k], vD, rDn[k])),
                            fmaxf(fmaf(pc[k], vL, rLt[k]),
                                  fmaf(pc[k], vR, rRt[k]))) - ro[k];
            float v = fmaxf(vs[k], m);
            v = ((vmask >> k) & 1u) ? v : 0.0f;
            vs[k] = v;
            if (t + k * 256 < NPOS) vout[i0] = v;
        }
        __syncthreads();
    }

    // Final v lives in the buffer written at it=9 -> sV1. Write inner 32x32 as
    // float4 (w0 and per-thread offset are multiples of 4 -> 16B aligned).
    {
        const int o  = t << 2;                  // 4 outputs per thread, 1024 total
        const int rr = o >> 5, cc = o & 31;     // within the 32x32 tile
        const float* src = &sV1[(HALO + rr) * REG + (HALO + cc)];
        float4 res;
        res.x = src[0]; res.y = src[1]; res.z = src[2]; res.w = src[3];
        *(float4*)&out[imgBase + (long)(h0 + rr) * IMG + (w0 + cc)] = res;
    }
}

extern "C" void kernel_launch(void* const* d_in, const int* in_sizes, int n_in,
                              void* d_out, int out_size, void* d_ws, size_t ws_size,
                              hipStream_t stream) {
    const float* obs = (const float*)d_in[0];   // (64,512,512,3) f32
    const float* Wp  = (const float*)d_in[1];   // (3,3) f32
    const float* bp  = (const float*)d_in[2];   // (3,) f32
    float* out = (float*)d_out;                 // (64,512,512) f32
    dim3 grid(IMG / TILE, IMG / TILE, 64);
    vin_kernel<<<grid, 256, 0, stream>>>(obs, Wp, bp, out);
}